// RGCN_65377992179803
// MI455X (gfx1250) — compile-verified
//
#include <hip/hip_runtime.h>

// RGCN, 2 layers, DIM=64, 8 relations, 100k nodes, 1.6M edges.
// Strategy (memory-bound, ~650MB HBM/layer @ 23.3TB/s => ~60us total):
//   1) cnt[dst*8+r] edge counts (identical for both layers -> computed once)
//   2) per layer: h[r] = x @ W[r] via V_WMMA_F32_16X16X4_F32 (9th "relation"
//      writes out = x @ Wroot + bias, initializing the output). W_r staged in
//      LDS once per workgroup; B fragments come from LDS (ds_load), A from
//      global (b64/lane).
//   3) per layer: out[dst] += h[etype][src] / max(cnt,1)  (f32 atomics into a
//      26MB destination -> L2-resident on MI455X's 192MB L2)

#define NN 100000
#define NE 1600000
#define FD 64
#define NR 8

typedef __attribute__((ext_vector_type(2))) float v2f;
typedef __attribute__((ext_vector_type(8))) float v8f;

__global__ void k_zero(float* __restrict__ p, int n) {
  int i = blockIdx.x * blockDim.x + threadIdx.x;
  if (i < n) p[i] = 0.0f;
}

__global__ void k_count(const int* __restrict__ ei, const int* __restrict__ et,
                        float* __restrict__ cnt) {
  int e = blockIdx.x * blockDim.x + threadIdx.x;
  if (e < NE) {
    int dst = ei[NE + e];
    int r   = et[e];
    atomicAdd(&cnt[dst * NR + r], 1.0f);
  }
}

// grid: (ceil(NN/128), 9). blockIdx.y = relation (8 == root transform).
// 256 threads = 8 waves; each wave computes a 16-node x 64-col tile.
__global__ void k_gemm(const float* __restrict__ x,
                       const float* __restrict__ W,      // [8][64][64] layer slice
                       const float* __restrict__ Wroot,  // [64][64]
                       const float* __restrict__ bias,   // [64]
                       float* __restrict__ h,            // [8][NN][64]
                       float* __restrict__ out) {        // [NN][64]
  __shared__ float sW[FD * FD];  // 16KB: this block's relation matrix

  const int tid = threadIdx.x;
  const int r   = blockIdx.y;   // 0..8
  const float* Wr = (r < NR) ? (W + (size_t)r * FD * FD) : Wroot;

  // cooperative stage of W_r into LDS (coalesced b128; 4 float4 per thread)
  {
    const float4* s = (const float4*)Wr;
    float4* d = (float4*)sW;
#pragma unroll
    for (int i = 0; i < 4; ++i) d[tid + 256 * i] = s[tid + 256 * i];
  }
  __syncthreads();  // all threads reach this before any wave-uniform early-out

  const int wave = tid >> 5;
  const int lane = tid & 31;
  const int half = lane >> 4;   // 0: lanes 0-15, 1: lanes 16-31
  const int lr   = lane & 15;
  const int m0   = blockIdx.x * 128 + wave * 16;
  if (m0 >= NN) return;         // wave-uniform: EXEC stays all-ones for WMMA

  const float* arow = x + (size_t)(m0 + lr) * FD;

  v8f acc0 = {}, acc1 = {}, acc2 = {}, acc3 = {};

  for (int k0 = 0; k0 < FD; k0 += 4) {
    const int kb = k0 + 2 * half;
    // A frag 16x4 f32: VGPR0 = K kb, VGPR1 = K kb+1 (per lane-half)
    v2f a;
    a.x = arow[kb];
    a.y = arow[kb + 1];
    // B frags 4x16 f32 for four 16-col tiles, from LDS
    const float* wk0 = sW + kb * FD;
    const float* wk1 = wk0 + FD;
    v2f b0, b1, b2, b3;
    b0.x = wk0[lr];      b0.y = wk1[lr];
    b1.x = wk0[16 + lr]; b1.y = wk1[16 + lr];
    b2.x = wk0[32 + lr]; b2.y = wk1[32 + lr];
    b3.x = wk0[48 + lr]; b3.y = wk1[48 + lr];
    acc0 = __builtin_amdgcn_wmma_f32_16x16x4_f32(false, a, false, b0, (short)0, acc0, false, false);
    acc1 = __builtin_amdgcn_wmma_f32_16x16x4_f32(false, a, false, b1, (short)0, acc1, false, false);
    acc2 = __builtin_amdgcn_wmma_f32_16x16x4_f32(false, a, false, b2, (short)0, acc2, false, false);
    acc3 = __builtin_amdgcn_wmma_f32_16x16x4_f32(false, a, false, b3, (short)0, acc3, false, false);
  }

  // D layout: VGPR v holds row m0 + v + 8*half, col c0 + lr
  if (r < NR) {
    float* hb = h + ((size_t)r * NN + m0) * FD + lr;
#pragma unroll
    for (int v = 0; v < 8; ++v) {
      float* p = hb + (size_t)(v + 8 * half) * FD;
      p[0]  = acc0[v];
      p[16] = acc1[v];
      p[32] = acc2[v];
      p[48] = acc3[v];
    }
  } else {
    const float bb0 = bias[lr], bb1 = bias[16 + lr], bb2 = bias[32 + lr], bb3 = bias[48 + lr];
    float* ob = out + (size_t)m0 * FD + lr;
#pragma unroll
    for (int v = 0; v < 8; ++v) {
      float* p = ob + (size_t)(v + 8 * half) * FD;
      p[0]  = acc0[v] + bb0;
      p[16] = acc1[v] + bb1;
      p[32] = acc2[v] + bb2;
      p[48] = acc3[v] + bb3;
    }
  }
}

// one thread per (edge, 4-float chunk): 25.6M threads, coalesced 256B/edge
__global__ void k_scatter(const float* __restrict__ h,
                          const int* __restrict__ ei,
                          const int* __restrict__ et,
                          const float* __restrict__ cnt,
                          float* __restrict__ out) {
  unsigned gid = blockIdx.x * blockDim.x + threadIdx.x;
  if (gid >= (unsigned)NE * 16u) return;
  const int e = (int)(gid >> 4);
  const int c = (int)(gid & 15u) * 4;
  const int src = ei[e];
  const int dst = ei[NE + e];
  const int r   = et[e];
  const float n = cnt[dst * NR + r];
  const float norm = 1.0f / fmaxf(n, 1.0f);
  const float4 m = *(const float4*)(h + ((size_t)r * NN + src) * FD + c);
  float* o = out + (size_t)dst * FD + c;
  atomicAdd(o + 0, m.x * norm);
  atomicAdd(o + 1, m.y * norm);
  atomicAdd(o + 2, m.z * norm);
  atomicAdd(o + 3, m.w * norm);
}

extern "C" void kernel_launch(void* const* d_in, const int* in_sizes, int n_in,
                              void* d_out, int out_size, void* d_ws, size_t ws_size,
                              hipStream_t stream) {
  const float* x      = (const float*)d_in[0];  // [NN][64]
  const float* weight = (const float*)d_in[1];  // [2][8][64][64]
  const float* root   = (const float*)d_in[2];  // [2][64][64]
  const float* bias   = (const float*)d_in[3];  // [2][64]
  const int*   ei     = (const int*)d_in[4];    // [2][NE]
  const int*   et     = (const int*)d_in[5];    // [NE]
  float* out = (float*)d_out;

  // workspace: cnt (3.2MB) | h (204.8MB) | x_tmp (25.6MB)
  char* ws = (char*)d_ws;
  float* cnt = (float*)ws;
  float* h   = (float*)(ws + (((size_t)NN * NR * 4 + 255) & ~(size_t)255));
  float* xt  = (float*)((char*)h + (size_t)NR * NN * FD * 4);

  const int nr_cnt = NN * NR;
  k_zero<<<(nr_cnt + 255) / 256, 256, 0, stream>>>(cnt, nr_cnt);
  k_count<<<(NE + 255) / 256, 256, 0, stream>>>(ei, et, cnt);

  dim3 ggrid((NN + 127) / 128, NR + 1);
  unsigned sblocks = ((unsigned)NE * 16u + 255u) / 256u;

  // layer 0: x -> xt
  k_gemm<<<ggrid, 256, 0, stream>>>(x, weight, root, bias, h, xt);
  k_scatter<<<sblocks, 256, 0, stream>>>(h, ei, et, cnt, xt);

  // layer 1: xt -> out
  k_gemm<<<ggrid, 256, 0, stream>>>(xt, weight + (size_t)NR * FD * FD,
                                    root + (size_t)FD * FD, bias + FD, h, out);
  k_scatter<<<sblocks, 256, 0, stream>>>(h, ei, et, cnt, out);
}